// Network_41764261986467
// MI455X (gfx1250) — compile-verified
//
#include <hip/hip_runtime.h>
#include <math.h>

typedef __attribute__((ext_vector_type(2))) float v2f;
typedef __attribute__((ext_vector_type(8))) float v8f;

#define B_TOT 65536
#define IN_D  16
#define HL_D  10
#define NL_D  3
#define T_D   200
#define NPOP_F 8600000.0f

__device__ __forceinline__ float softplusf(float x) {
    // numerically stable softplus: max(x,0) + log1p(exp(-|x|))
    return fmaxf(x, 0.0f) + log1pf(expf(-fabsf(x)));
}

// ---------------------------------------------------------------------------
// Kernel 1: batched tiny-MLP evaluation with fp32 WMMA (16x16x4).
//
// Each wave owns one tile of 16 batch samples; hidden dim 10 is padded to 16
// with zeros (padded units stay exactly 0 through tanh since weights, bias and
// inputs are all zero there). K=16 is accumulated as 4 chained WMMAs.
//
// All weights/biases are staged ONCE per block into LDS in padded [N][K]
// B-matrix layout, so every B fragment is an unconditional 8-byte-aligned
// ds_load — no divergent global loads in the inner pipeline.
// C-layout -> A-layout relayout between layers goes through a per-wave padded
// LDS tile.
// ---------------------------------------------------------------------------
__global__ __launch_bounds__(128) void mlp_wmma_kernel(
    const float* __restrict__ data,   // [B,16]
    const float* __restrict__ W0,     // [3,10,16]
    const float* __restrict__ b0,     // [3,10]
    const float* __restrict__ Wh,     // [3,3,10,10]
    const float* __restrict__ bh,     // [3,3,10]
    const float* __restrict__ Wo,     // [3,1,10]
    const float* __restrict__ bo,     // [3,1]
    float* __restrict__ vals)         // [3,B]  (gamma, beta, I0)
{
    __shared__ float tile[4][16][17];        // per-wave 16x16 h tile, padded
    __shared__ float w0p[3][16][16];         // [n][N][K]   padded W0
    __shared__ float whp[3][NL_D][16][16];   // [n][l][N][K] padded Wh
    __shared__ float wop[3][16];             // [n][K]      padded Wo
    __shared__ float b0p[3][16];             // [n][N]      padded b0
    __shared__ float bhp[3][NL_D][16];       // [n][l][N]   padded bh
    __shared__ float bop[3];                 // [n]

    const int tid = threadIdx.x;

    // ------- one-time weight staging (branch-free: clamp index, then select)
    for (int e = tid; e < 3 * 16 * 16; e += 128) {            // W0 pad
        const int n = e >> 8, N = (e >> 4) & 15, K = e & 15;
        const float v = W0[(n * HL_D + min(N, HL_D - 1)) * IN_D + K];
        w0p[n][N][K] = (N < HL_D) ? v : 0.0f;
    }
    for (int e = tid; e < 3 * NL_D * 16 * 16; e += 128) {     // Wh pad
        const int n = e >> 10, l = (e >> 8) & 3, N = (e >> 4) & 15, K = e & 15;
        const float v = Wh[((n * NL_D + l) * HL_D + min(N, HL_D - 1)) * HL_D +
                           min(K, HL_D - 1)];
        whp[n][l][N][K] = (N < HL_D && K < HL_D) ? v : 0.0f;
    }
    for (int e = tid; e < 3 * 16; e += 128) {                 // Wo + b0 pad
        const int n = e >> 4, K = e & 15;
        const float wv = Wo[n * HL_D + min(K, HL_D - 1)];
        const float bv = b0[n * HL_D + min(K, HL_D - 1)];
        wop[n][K] = (K < HL_D) ? wv : 0.0f;
        b0p[n][K] = (K < HL_D) ? bv : 0.0f;
    }
    for (int e = tid; e < 3 * NL_D * 16; e += 128) {          // bh pad
        const int n = e >> 6, l = (e >> 4) & 3, N = e & 15;
        const float v = bh[(n * NL_D + l) * HL_D + min(N, HL_D - 1)];
        bhp[n][l][N] = (N < HL_D) ? v : 0.0f;
    }
    if (tid < 3) bop[tid] = bo[tid];
    __syncthreads();

    // ------- per-wave tile setup
    const int lane = tid & 31;
    const int wave = tid >> 5;
    const int base = (blockIdx.x * 4 + wave) * 16;

    const int r  = lane & 15;                 // M (for A) or N (for B/C)
    const int kh = (lane >> 4) * 2;           // K sub-offset within a chunk
    const int mh = (lane >> 4) * 8;           // M offset for C rows

    // A fragments for the input tile x[16 x 16] (shared across the 3 nets).
    v2f afr[4];
#pragma unroll
    for (int c = 0; c < 4; ++c)
        afr[c] = *(const v2f*)(data + (base + r) * IN_D + (c * 4 + kh));

    for (int n = 0; n < 3; ++n) {
        // ---- layer 0: h = tanh(x @ W0[n]^T + b0[n])
        v8f acc = {0.f, 0.f, 0.f, 0.f, 0.f, 0.f, 0.f, 0.f};
#pragma unroll
        for (int c = 0; c < 4; ++c) {
            const v2f bfr = *(const v2f*)&w0p[n][r][c * 4 + kh];
            acc = __builtin_amdgcn_wmma_f32_16x16x4_f32(
                false, afr[c], false, bfr, (short)0, acc, false, false);
        }
        {
            const float bias = b0p[n][r];
#pragma unroll
            for (int j = 0; j < 8; ++j)
                tile[wave][j + mh][r] = tanhf(acc[j] + bias);
        }
        __syncthreads();

        // ---- hidden layers: h = tanh(h @ Wh[n][l]^T + bh[n][l])
        for (int l = 0; l < NL_D; ++l) {
            v8f hacc = {0.f, 0.f, 0.f, 0.f, 0.f, 0.f, 0.f, 0.f};
#pragma unroll
            for (int c = 0; c < 4; ++c) {
                const int k = c * 4 + kh;
                v2f a;
                a.x = tile[wave][r][k];
                a.y = tile[wave][r][k + 1];
                const v2f bfr = *(const v2f*)&whp[n][l][r][k];
                hacc = __builtin_amdgcn_wmma_f32_16x16x4_f32(
                    false, a, false, bfr, (short)0, hacc, false, false);
            }
            __syncthreads();   // all reads of the tile done before overwrite
            {
                const float bias = bhp[n][l][r];
#pragma unroll
                for (int j = 0; j < 8; ++j)
                    tile[wave][j + mh][r] = tanhf(hacc[j] + bias);
            }
            __syncthreads();
        }

        // ---- output layer: out = h @ Wo[n]^T + bo[n]   (single column N=0)
        v8f oacc = {0.f, 0.f, 0.f, 0.f, 0.f, 0.f, 0.f, 0.f};
#pragma unroll
        for (int c = 0; c < 4; ++c) {
            const int k = c * 4 + kh;
            v2f a;
            a.x = tile[wave][r][k];
            a.y = tile[wave][r][k + 1];
            // unconditional LDS loads + cndmask select (no branch)
            v2f bfr;
            bfr.x = (r == 0) ? wop[n][k]     : 0.0f;
            bfr.y = (r == 0) ? wop[n][k + 1] : 0.0f;
            oacc = __builtin_amdgcn_wmma_f32_16x16x4_f32(
                false, a, false, bfr, (short)0, oacc, false, false);
        }
        if (r == 0) {          // lanes 0 and 16 hold column N=0
            const float ob = bop[n];
#pragma unroll
            for (int j = 0; j < 8; ++j)
                vals[n * B_TOT + base + j + mh] = softplusf(oacc[j] + ob);
        }
        __syncthreads();       // tile reused by next net
    }
}

// ---------------------------------------------------------------------------
// Kernel 2: per-sample RK4 integration of the SIR ODE (S and I only; R never
// feeds back). Coalesced out[t*B + b] stores each step: 52.4 MB total — this
// is the bandwidth-bound part of the whole problem (~2.5 us floor at 23.3 TB/s).
// ---------------------------------------------------------------------------
__global__ __launch_bounds__(256) void sir_rk4_kernel(
    const float* __restrict__ vals,   // [3,B]
    const float* __restrict__ times,  // [T]
    float* __restrict__ out)          // [T*B]
{
    const int b = blockIdx.x * blockDim.x + threadIdx.x;
    if (b >= B_TOT) return;

    const float gamma = vals[0 * B_TOT + b];
    const float beta  = vals[1 * B_TOT + b];
    const float I0    = vals[2 * B_TOT + b];

    float S = NPOP_F - I0;
    float I = I0;
    out[b] = I;                        // t = 0

    float tprev = times[0];
    for (int t = 0; t < T_D - 1; ++t) {
        const float tcur = times[t + 1];
        const float dt = tcur - tprev;
        tprev = tcur;

        const float f1 = beta * S * I / NPOP_F;
        const float k1S = -f1,                k1I = f1 - gamma * I;

        const float S2 = S + 0.5f * dt * k1S, I2 = I + 0.5f * dt * k1I;
        const float f2 = beta * S2 * I2 / NPOP_F;
        const float k2S = -f2,                k2I = f2 - gamma * I2;

        const float S3 = S + 0.5f * dt * k2S, I3 = I + 0.5f * dt * k2I;
        const float f3 = beta * S3 * I3 / NPOP_F;
        const float k3S = -f3,                k3I = f3 - gamma * I3;

        const float S4 = S + dt * k3S,        I4 = I + dt * k3I;
        const float f4 = beta * S4 * I4 / NPOP_F;
        const float k4S = -f4,                k4I = f4 - gamma * I4;

        const float c = dt * (1.0f / 6.0f);
        S += c * (k1S + 2.0f * k2S + 2.0f * k3S + k4S);
        I += c * (k1I + 2.0f * k2I + 2.0f * k3I + k4I);

        out[(t + 1) * B_TOT + b] = I;
    }
}

extern "C" void kernel_launch(void* const* d_in, const int* in_sizes, int n_in,
                              void* d_out, int out_size, void* d_ws, size_t ws_size,
                              hipStream_t stream) {
    const float* data  = (const float*)d_in[0];   // [B,16]
    const float* times = (const float*)d_in[1];   // [T]
    const float* W0    = (const float*)d_in[2];   // [3,10,16]
    const float* b0    = (const float*)d_in[3];   // [3,10]
    const float* Wh    = (const float*)d_in[4];   // [3,3,10,10]
    const float* bh    = (const float*)d_in[5];   // [3,3,10]
    const float* Wo    = (const float*)d_in[6];   // [3,1,10]
    const float* bo    = (const float*)d_in[7];   // [3,1]

    float* vals = (float*)d_ws;                   // 3*B floats scratch
    float* out  = (float*)d_out;                  // [T*B]

    // 4096 wave-tiles of 16 samples, 4 waves (128 threads) per block
    mlp_wmma_kernel<<<dim3(B_TOT / 16 / 4), dim3(128), 0, stream>>>(
        data, W0, b0, Wh, bh, Wo, bo, vals);

    sir_rk4_kernel<<<dim3((B_TOT + 255) / 256), dim3(256), 0, stream>>>(
        vals, times, out);
}